// myHippo_13022340841659
// MI455X (gfx1250) — compile-verified
//
#include <hip/hip_runtime.h>
#include <stdint.h>

// ---------------------------------------------------------------------------
// memPool cosine-similarity recall + in-place pool update, fused single pass.
//
// MI455X roofline: 0.7 GFLOP vs 0.52 GB traffic -> memory bound (~22us floor
// at 23.3 TB/s). Hot loop streams memPool once via CDNA5 async global->LDS
// (ASYNCcnt) with double buffering; the out-vector partial reduction is a
// genuine ones^T @ P matmul and uses V_WMMA_F32_16X16X4_F32.
// ---------------------------------------------------------------------------

typedef float f4  __attribute__((ext_vector_type(4)));
typedef float v2f __attribute__((ext_vector_type(2)));
typedef float v8f __attribute__((ext_vector_type(8)));

#define POOL 65536
#define DIM  1024
#define TPB  256
#define BBLK 1024              // blocks in kmain
#define RPB  (POOL / BBLK)     // 64 rows per block
#define EPSI 1e-8f

// scratch layout in floats
#define WS_XN   0              // [0]       : ||x||
#define WS_OUTU 64             // [64..1087]: un-normalized out
#define WS_PART 2048           // [2048 + ) : BBLK x DIM partial out vectors (4MB)

// ---- CDNA5 async global->LDS (GVS mode: SGPR base + per-lane i32 offset) ----
__device__ __forceinline__ void async_load16(uint32_t ldsAddr, uint64_t base, uint32_t goff) {
  asm volatile("global_load_async_to_lds_b128 %0, %1, %2"
               :: "v"(ldsAddr), "v"(goff), "s"(base) : "memory");
}
__device__ __forceinline__ void wait_async_le1() {
  asm volatile("s_wait_asynccnt 0x1" ::: "memory");
}
__device__ __forceinline__ void wait_async_le0() {
  asm volatile("s_wait_asynccnt 0x0" ::: "memory");
}

// ---- block reductions (wave32 shfl + LDS), deterministic order ----
__device__ __forceinline__ float2 blockReduceSum2(float a, float b, float* l) {
  for (int m = 16; m; m >>= 1) { a += __shfl_xor(a, m, 32); b += __shfl_xor(b, m, 32); }
  int w = threadIdx.x >> 5;
  if ((threadIdx.x & 31) == 0) { l[w] = a; l[8 + w] = b; }
  __syncthreads();
  if (threadIdx.x == 0) {
    float sa = 0.f, sb = 0.f;
    for (int i = 0; i < 8; ++i) { sa += l[i]; sb += l[8 + i]; }
    l[16] = sa; l[17] = sb;
  }
  __syncthreads();
  float2 r; r.x = l[16]; r.y = l[17];
  return r;
}

__device__ __forceinline__ float blockReduceMax(float a, float* l) {
  for (int m = 16; m; m >>= 1) a = fmaxf(a, __shfl_xor(a, m, 32));
  int w = threadIdx.x >> 5;
  if ((threadIdx.x & 31) == 0) l[w] = a;
  __syncthreads();
  if (threadIdx.x == 0) {
    float s = l[0];
    for (int i = 1; i < 8; ++i) s = fmaxf(s, l[i]);
    l[16] = s;
  }
  __syncthreads();
  return l[16];
}

// ---- kernel 1: ||x|| ----
__global__ void __launch_bounds__(TPB) knorm(const float* __restrict__ x, float* __restrict__ W) {
  __shared__ float red[18];
  int t = threadIdx.x;
  f4 v = ((const f4*)x)[t];
  float s = v.x * v.x + v.y * v.y + v.z * v.z + v.w * v.w;
  float2 r = blockReduceSum2(s, 0.f, red);
  if (t == 0) W[WS_XN] = fmaxf(sqrtf(r.x), EPSI);
}

// ---- kernel 2: fused sim / out-partials / pool update, async double-buffer ----
__global__ void __launch_bounds__(TPB, 1)
kmain(const float* __restrict__ pool, const float* __restrict__ x,
      float* __restrict__ newPool, float* __restrict__ W) {
  __shared__ float red[18];
  __shared__ __align__(16) float buf[2][DIM];   // 8 KB double buffer

  const int t = threadIdx.x;
  const int row0 = blockIdx.x * RPB;
  const float xn = W[WS_XN];
  const f4 xv = ((const f4*)x)[t];
  const uint64_t base = (uint64_t)pool;
  const uint32_t lds0 = (uint32_t)(uintptr_t)(&buf[0][0]);

  f4 acc = {0.f, 0.f, 0.f, 0.f};

  // prologue: stage row0 into buf[0]  (each lane moves 16B; 256 lanes = 4KB row)
  async_load16(lds0 + (uint32_t)t * 16u,
               base, (uint32_t)row0 * (DIM * 4u) + (uint32_t)t * 16u);

  for (int r = 0; r < RPB; ++r) {
    const int cur = r & 1;
    if (r + 1 < RPB) {
      async_load16(lds0 + (uint32_t)(cur ^ 1) * (DIM * 4u) + (uint32_t)t * 16u,
                   base, (uint32_t)(row0 + r + 1) * (DIM * 4u) + (uint32_t)t * 16u);
      wait_async_le1();               // oldest (buf[cur]) complete
    } else {
      wait_async_le0();
    }
    __syncthreads();                  // buf[cur] visible to all waves

    f4 v = ((const f4*)buf[cur])[t];
    float d = v.x * xv.x + v.y * xv.y + v.z * xv.z + v.w * xv.w;
    float n = v.x * v.x + v.y * v.y + v.z * v.z + v.w * v.w;
    float2 s = blockReduceSum2(d, n, red);
    float sim = s.x / (fmaxf(sqrtf(s.y), EPSI) * xn);

    acc += sim * v;                   // out partial uses PRE-update row

    f4 u = v + sim * xv;
    float am = fmaxf(fmaxf(fabsf(u.x), fabsf(u.y)), fmaxf(fabsf(u.z), fabsf(u.w)));
    float m = blockReduceMax(am, red);
    f4 o = (m != 0.f) ? (u / m) : u;
    __builtin_nontemporal_store(o, (f4*)(newPool + (size_t)(row0 + r) * DIM) + t);

    __syncthreads();                  // all reads of buf[cur] done before reuse
  }

  ((f4*)(W + WS_PART))[blockIdx.x * (DIM / 4) + t] = acc;
}

// ---- kernel 3: out_unnorm[c] = sum_p PART[p][c] via V_WMMA_F32_16X16X4_F32 ----
// A = all-ones 16x4 -> D[m][n] = column sums of B (invariant to K-slot layout).
__global__ void __launch_bounds__(TPB) kreduce(const float* __restrict__ P,
                                               float* __restrict__ W) {
  __shared__ float l[8 * 16];
  const int t = threadIdx.x;
  const int wv = t >> 5;              // wave 0..7
  const int lane = t & 31;
  const int n = lane & 15;            // column within 16-col tile
  const int kb = (lane >> 4) ? 2 : 0; // K pair held by this half-wave
  const int col0 = blockIdx.x * 16;

  const v2f ones = {1.f, 1.f};
  v8f acc = {0.f, 0.f, 0.f, 0.f, 0.f, 0.f, 0.f, 0.f};

  const int p0 = wv * (BBLK / 8);     // 128 partial rows per wave
  for (int p = p0; p < p0 + BBLK / 8; p += 4) {
    v2f bf;
    bf.x = P[(size_t)(p + kb) * DIM + col0 + n];
    bf.y = P[(size_t)(p + kb + 1) * DIM + col0 + n];
    acc = __builtin_amdgcn_wmma_f32_16x16x4_f32(false, ones, false, bf,
                                                (short)0, acc, false, false);
  }
  // D row M=0: lanes 0..15, VGPR0 hold column sums for N=0..15
  if (lane < 16) l[wv * 16 + n] = acc[0];
  __syncthreads();
  if (t < 16) {
    float s = 0.f;
    for (int w = 0; w < 8; ++w) s += l[w * 16 + t];
    W[WS_OUTU + col0 + t] = s;
  }
}

// ---- kernel 4: out = out_unnorm / max|out_unnorm| ----
__global__ void __launch_bounds__(TPB) kfinal(const float* __restrict__ W,
                                              float* __restrict__ out) {
  __shared__ float red[18];
  int t = threadIdx.x;
  f4 v = ((const f4*)(W + WS_OUTU))[t];
  float am = fmaxf(fmaxf(fabsf(v.x), fabsf(v.y)), fmaxf(fabsf(v.z), fabsf(v.w)));
  float m = blockReduceMax(am, red);
  ((f4*)out)[t] = v / m;
}

extern "C" void kernel_launch(void* const* d_in, const int* in_sizes, int n_in,
                              void* d_out, int out_size, void* d_ws, size_t ws_size,
                              hipStream_t stream) {
  (void)in_sizes; (void)n_in; (void)out_size; (void)ws_size;
  const float* x    = (const float*)d_in[0];   // [1024]
  const float* pool = (const float*)d_in[1];   // [65536 x 1024]
  float* out     = (float*)d_out;              // [1024] ++ [65536 x 1024]
  float* newPool = out + DIM;
  float* W       = (float*)d_ws;               // needs ~4.2 MB scratch

  knorm  <<<1,        TPB, 0, stream>>>(x, W);
  kmain  <<<BBLK,     TPB, 0, stream>>>(pool, x, newPool, W);
  kreduce<<<DIM / 16, TPB, 0, stream>>>(W + WS_PART, W);
  kfinal <<<1,        TPB, 0, stream>>>(W, out);
}